// SACCOMACriticNS_84911503442085
// MI455X (gfx1250) — compile-verified
//
#include <hip/hip_runtime.h>

// ---------------------------------------------------------------------------
// DGCNN-style critic, fully fused: one workgroup (256 thr = 8 waves) per
// point cloud. All GEMMs on V_WMMA_F32_16X16X4_F32 (exact fp32).
// All activations live in LDS, point-major [n][c] with bank-conflict-free
// padded strides so WMMA fragments are contiguous b64 loads / b128 stores.
// ---------------------------------------------------------------------------

typedef float v2f __attribute__((ext_vector_type(2)));
typedef float v8f __attribute__((ext_vector_type(8)));

#define NPTS     100
#define KNN      10
#define CIN      30
#define NEG      0.2f
#define NTHREADS 256
#define NWAVES   8

// ---- LDS layout (float indices); all strides keep (stride*lane)%64 distinct
constexpr int F_IDX = 0;                  // 1000 idx stored as int-bits
constexpr int F_X0  = 1024;               // x0:  [112][36]
constexpr int F_XST = F_X0 + 112 * 36;    // 5056: [112][100]  x1|x2|x3 (+pad)
constexpr int F_U   = F_XST + 112 * 100;  // 16256: u [112][36]
constexpr int F_V   = F_U + 112 * 36;     // 20288: v [112][36]
constexpr int F_WA  = F_V + 112 * 36;     // 24320: 3 staged 32x36 weight bufs
constexpr int F_GV  = F_WA + 3 * 32 * 36; // 27776: 512 global feature
constexpr int F_GO  = F_GV + 512;         // 28288: 128 (W7@g), also xx scratch
constexpr int F_SCR = F_GO + 128;         // 28416: scratch 14784 f
                                          //   xin[112][36] / gram[112][116]
                                          //   yc[160][36]+zc[160][36] / h1[112][132]
constexpr int F_TOT = F_SCR + 14784;      // 43200 floats
#define SMEM_BYTES (F_TOT * 4)            // 172800 B

__device__ __forceinline__ float lrelu(float x) { return x > 0.f ? x : NEG * x; }

__device__ __forceinline__ float  ldsf(int i) { extern __shared__ float lds[]; return lds[i]; }
__device__ __forceinline__ void   stsf(int i, float v) { extern __shared__ float lds[]; lds[i] = v; }
__device__ __forceinline__ v2f    lds2(int i) { extern __shared__ float lds[]; return *(v2f*)&lds[i]; }
__device__ __forceinline__ void   sts2(int i, v2f v) { extern __shared__ float lds[]; *(v2f*)&lds[i] = v; }
__device__ __forceinline__ float4 lds4(int i) { extern __shared__ float lds[]; return *(float4*)&lds[i]; }
__device__ __forceinline__ void   sts4(int i, float4 v) { extern __shared__ float lds[]; *(float4*)&lds[i] = v; }

// ---- 16x16 tile GEMM on V_WMMA_F32_16X16X4_F32, K fully unrolled ----------
// loaders get k-base (kk*4); lane's 2*hi sub-offset is folded into the base.
template <int KS, typename FA, typename FB>
__device__ __forceinline__ v8f gemm_tile(FA la, FB lb) {
  v8f acc = {0.f, 0.f, 0.f, 0.f, 0.f, 0.f, 0.f, 0.f};
#pragma unroll
  for (int kk = 0; kk < KS; ++kk)
    acc = __builtin_amdgcn_wmma_f32_16x16x4_f32(false, la(kk << 2), false,
                                                lb(kk << 2), (short)0, acc,
                                                false, false);
  return acc;
}

// raw store of the 8 consecutive channels a lane owns (16B aligned)
__device__ __forceinline__ void st_tile_raw(int dst, const v8f& a) {
  sts4(dst, make_float4(a[0], a[1], a[2], a[3]));
  sts4(dst + 4, make_float4(a[4], a[5], a[6], a[7]));
}

// bn + leaky-relu store, channels obase..obase+7 (obase multiple of 8)
__device__ __forceinline__ void st_tile_bn(int dst, int obase,
                                           const float* __restrict__ s,
                                           const float* __restrict__ t,
                                           const v8f& a) {
  float4 sa = *(const float4*)&s[obase], sb = *(const float4*)&s[obase + 4];
  float4 ta = *(const float4*)&t[obase], tb = *(const float4*)&t[obase + 4];
  sts4(dst, make_float4(lrelu(sa.x * a[0] + ta.x), lrelu(sa.y * a[1] + ta.y),
                        lrelu(sa.z * a[2] + ta.z), lrelu(sa.w * a[3] + ta.w)));
  sts4(dst + 4,
       make_float4(lrelu(sb.x * a[4] + tb.x), lrelu(sb.y * a[5] + tb.y),
                   lrelu(sb.z * a[6] + tb.z), lrelu(sb.w * a[7] + tb.w)));
}

// ---- kNN: Gram matrix via WMMA (symmetric -> contiguous store), top-10 ----
__device__ void knn_topk(int xsBase, int xsStr) {
  const int tid = threadIdx.x, wave = tid >> 5;
  const int lane = tid & 31, lo = lane & 15, hi = lane >> 4, khi = hi << 1;
  {
    int tm = 0, tn = wave;
    while (tn >= 7) { tn -= 7; ++tm; }
    for (int t = wave; t < 49; t += NWAVES) {
      const int mo = tm << 4, no = tn << 4;
      const int ab = xsBase + (mo + lo) * xsStr + khi;
      const int bb = xsBase + (no + lo) * xsStr + khi;
      v8f acc = gemm_tile<8>([&](int k) { return lds2(ab + k); },
                             [&](int k) { return lds2(bb + k); });
      st_tile_raw(F_SCR + (no + lo) * 116 + mo + (hi << 3), acc);
      tn += NWAVES;
      while (tn >= 7) { tn -= 7; ++tm; }
    }
  }
  __syncthreads();
  if (tid < NPTS) stsf(F_GO + tid, ldsf(F_SCR + tid * 117));  // diag = |x|^2
  __syncthreads();
  if (tid < NPTS) {
    const int n = tid;
    const float dn = ldsf(F_GO + n);
    float bv[KNN];
    int bi[KNN];
#pragma unroll
    for (int j = 0; j < KNN; ++j) { bv[j] = -3.0e38f; bi[j] = 0; }
    for (int m = 0; m < NPTS; ++m) {
      const float d = 2.f * ldsf(F_SCR + n * 116 + m) - dn - ldsf(F_GO + m);
      if (d > bv[KNN - 1]) {
        int p = KNN - 1;
#pragma unroll
        for (int j = KNN - 2; j >= 0; --j)
          if (bv[j] < d) { bv[j + 1] = bv[j]; bi[j + 1] = bi[j]; p = j; }
        bv[p] = d;
        bi[p] = m;
      }
    }
#pragma unroll
    for (int j = 0; j < KNN; ++j)
      stsf(F_IDX + n * KNN + j, __int_as_float(bi[j]));
  }
  __syncthreads();
}

// ---- one edge-conv round --------------------------------------------------
// y[o,n,k] = lrelu(s1*( (Wa@x)[o,idx[n,k]] + ((Wb-Wa)@x)[o,n] ) + t1)
// optional W2 (32x32)+bn2, then max over k -> xo[n][c]
__device__ void edge_round(const float* __restrict__ We,
                           const float* __restrict__ s1,
                           const float* __restrict__ t1,
                           const float* __restrict__ W2p,
                           const float* __restrict__ s2,
                           const float* __restrict__ t2, int xsBase, int xsStr,
                           int xoBase, int xoStr) {
  const int tid = threadIdx.x, wave = tid >> 5;
  const int lane = tid & 31, lo = lane & 15, hi = lane >> 4, khi = hi << 1;
  // stage Wa -> cV, (Wb-Wa) -> cU, W2 -> cW2 (strided 36, div-free indexing)
  for (int i = tid; i < 32 * 32; i += NTHREADS) {
    const int m = i >> 5, k = i & 31;
    const float wa = We[(m << 6) + k];
    stsf(F_WA + m * 36 + k, wa);
    stsf(F_WA + 1152 + m * 36 + k, We[(m << 6) + 32 + k] - wa);
    if (W2p) stsf(F_WA + 2304 + m * 36 + k, W2p[i]);
  }
  __syncthreads();
  // v = Wa@xs ; u = (Wb-Wa)@xs  (tile grid 4x7: rows 0,1 -> v ; 2,3 -> u)
  {
    int tm = 0, tn = wave;
    while (tn >= 7) { tn -= 7; ++tm; }
    for (int t = wave; t < 28; t += NWAVES) {
      const int which = tm >> 1;
      const int mo = (tm & 1) << 4, no = tn << 4;
      const int ab = F_WA + which * 1152 + (mo + lo) * 36 + khi;
      const int bb = xsBase + (no + lo) * xsStr + khi;
      v8f acc = gemm_tile<8>([&](int k) { return lds2(ab + k); },
                             [&](int k) { return lds2(bb + k); });
      st_tile_raw((which ? F_U : F_V) + (no + lo) * 36 + mo + (hi << 3), acc);
      tn += NWAVES;
      while (tn >= 7) { tn -= 7; ++tm; }
    }
  }
  __syncthreads();

  for (int p0 = 0; p0 < NPTS; p0 += 16) {  // 16 points = 160 edges per chunk
    if (tid < 160) {
      const int pl = tid / 10, kk = tid - pl * 10;
      int pp = p0 + pl;
      if (pp > NPTS - 1) pp = NPTS - 1;
      const int j = __float_as_int(ldsf(F_IDX + pp * KNN + kk));
      const int vb = F_V + j * 36, ub = F_U + pp * 36, yb = F_SCR + tid * 36;
#pragma unroll
      for (int c = 0; c < 32; c += 4) {
        float4 vv = lds4(vb + c), uu = lds4(ub + c);
        float4 ss = *(const float4*)&s1[c], tt = *(const float4*)&t1[c];
        sts4(yb + c,
             make_float4(lrelu(ss.x * (vv.x + uu.x) + tt.x),
                         lrelu(ss.y * (vv.y + uu.y) + tt.y),
                         lrelu(ss.z * (vv.z + uu.z) + tt.z),
                         lrelu(ss.w * (vv.w + uu.w) + tt.w)));
      }
    }
    __syncthreads();
    int srcb = F_SCR;
    if (W2p) {
      int tm = 0, tn = wave;
      while (tn >= 10) { tn -= 10; ++tm; }
      for (int t = wave; t < 20; t += NWAVES) {
        const int mo = tm << 4, no = tn << 4;
        const int ab = F_WA + 2304 + (mo + lo) * 36 + khi;
        const int bb = F_SCR + (no + lo) * 36 + khi;
        v8f acc = gemm_tile<8>([&](int k) { return lds2(ab + k); },
                               [&](int k) { return lds2(bb + k); });
        const int ob = mo + (hi << 3);
        st_tile_bn(F_SCR + 5760 + (no + lo) * 36 + ob, ob, s2, t2, acc);
        tn += NWAVES;
        while (tn >= 10) { tn -= 10; ++tm; }
      }
      __syncthreads();
      srcb = F_SCR + 5760;
    }
    for (int i = tid; i < 512; i += NTHREADS) {  // 32 ch x 16 points
      const int c = i & 31, pl = i >> 5, pp = p0 + pl;
      if (pp < NPTS) {
        float m = -3.0e38f;
        const int base = srcb + pl * 360 + c;
#pragma unroll
        for (int kk = 0; kk < KNN; ++kk) m = fmaxf(m, ldsf(base + kk * 36));
        stsf(xoBase + pp * xoStr + c, m);
      }
    }
    __syncthreads();
  }
}

struct Params {
  const float* obs;
  const float* W[10];
  const float* s[9];
  const float* t[9];
  float* out;
};

__global__ __launch_bounds__(NTHREADS) void dgcnn_kernel(Params p) {
  extern __shared__ float lds[];
  const int tid = threadIdx.x, wave = tid >> 5;
  const int lane = tid & 31, lo = lane & 15, hi = lane >> 4, khi = hi << 1;
  const int b = blockIdx.x;
  const float* obs = p.obs + (size_t)b * (NPTS * CIN);
  __builtin_prefetch(obs, 0, 1);

  // zero all LDS once (provides every pad region)
  for (int i = tid * 4; i < F_TOT; i += NTHREADS * 4)
    sts4(i, make_float4(0.f, 0.f, 0.f, 0.f));
  __syncthreads();

  // stage obs: xin[n][36] (c<30), and W0 -> cW0[m][36] zero-padded K 30->32
  if (tid < NPTS) {
    const float* src = obs + tid * CIN;
    const int basei = F_SCR + tid * 36;
#pragma unroll
    for (int c = 0; c < CIN; c += 2) sts2(basei + c, *(const v2f*)&src[c]);
  }
  for (int i = tid; i < 32 * 32; i += NTHREADS) {
    const int m = i >> 5, k = i & 31;
    stsf(F_WA + m * 36 + k, (k < CIN) ? p.W[0][m * CIN + k] : 0.f);
  }
  __syncthreads();

  // conv0: x0[n][c] = lrelu(s0*(W0 @ xin)+t0), pad rows -> 0
  {
    const float* s0 = p.s[0];
    const float* t0 = p.t[0];
    int tm = 0, tn = wave;
    while (tn >= 7) { tn -= 7; ++tm; }
    for (int t = wave; t < 14; t += NWAVES) {
      const int mo = tm << 4, no = tn << 4;
      const int ab = F_WA + (mo + lo) * 36 + khi;
      const int bb = F_SCR + (no + lo) * 36 + khi;
      v8f acc = gemm_tile<8>([&](int k) { return lds2(ab + k); },
                             [&](int k) { return lds2(bb + k); });
      const int n = no + lo, ob = mo + (hi << 3);
      const int dst = F_X0 + n * 36 + ob;
      if (n < NPTS) {
        st_tile_bn(dst, ob, s0, t0, acc);
      } else {
        sts4(dst, make_float4(0.f, 0.f, 0.f, 0.f));
        sts4(dst + 4, make_float4(0.f, 0.f, 0.f, 0.f));
      }
      tn += NWAVES;
      while (tn >= 7) { tn -= 7; ++tm; }
    }
  }
  __syncthreads();

  // ---- three edge-conv rounds (x1|x2|x3 land in xst columns 0/32/64) -----
  knn_topk(F_X0, 36);
  edge_round(p.W[1], p.s[1], p.t[1], p.W[2], p.s[2], p.t[2], F_X0, 36,
             F_XST + 0, 100);
  knn_topk(F_XST + 0, 100);
  edge_round(p.W[3], p.s[3], p.t[3], p.W[4], p.s[4], p.t[4], F_XST + 0, 100,
             F_XST + 32, 100);
  knn_topk(F_XST + 32, 100);
  edge_round(p.W[5], p.s[5], p.t[5], nullptr, nullptr, nullptr, F_XST + 32,
             100, F_XST + 64, 100);

  // ---- W6 (512x96) + max over n, folded per wave-owned tile row ----------
  {
    const float* W6 = p.W[6];
    const float* s6 = p.s[6];
    const float* t6 = p.t[6];
    for (int tm = wave; tm < 32; tm += NWAVES) {
      const int mo = tm << 4, ob = mo + (hi << 3);
      const float* arow = W6 + (mo + lo) * 96 + khi;
      float4 sa = *(const float4*)&s6[ob], sb = *(const float4*)&s6[ob + 4];
      float4 ta = *(const float4*)&t6[ob], tb = *(const float4*)&t6[ob + 4];
      const float sv[8] = {sa.x, sa.y, sa.z, sa.w, sb.x, sb.y, sb.z, sb.w};
      const float tv[8] = {ta.x, ta.y, ta.z, ta.w, tb.x, tb.y, tb.z, tb.w};
      float lmax[8];
#pragma unroll
      for (int r = 0; r < 8; ++r) lmax[r] = -3.0e38f;
      for (int tn = 0; tn < 7; ++tn) {
        const int no = tn << 4;
        const int bb = F_XST + (no + lo) * 100 + khi;
        v8f acc =
            gemm_tile<24>([&](int k) { return *(const v2f*)&arow[k]; },
                          [&](int k) { return lds2(bb + k); });
        if (no + lo < NPTS) {
#pragma unroll
          for (int r = 0; r < 8; ++r)
            lmax[r] = fmaxf(lmax[r], lrelu(sv[r] * acc[r] + tv[r]));
        }
      }
#pragma unroll
      for (int r = 0; r < 8; ++r) {  // reduce across the 16 column-lanes
        float mv = lmax[r];
        mv = fmaxf(mv, __shfl_xor(mv, 1, 32));
        mv = fmaxf(mv, __shfl_xor(mv, 2, 32));
        mv = fmaxf(mv, __shfl_xor(mv, 4, 32));
        mv = fmaxf(mv, __shfl_xor(mv, 8, 32));
        lmax[r] = mv;
      }
      if (lo == 0)
#pragma unroll
        for (int r = 0; r < 8; ++r) stsf(F_GV + ob + r, lmax[r]);
    }
  }
  __syncthreads();

  // ---- go = W7[:, :512] @ g  (broadcast part of the concat) --------------
  if (tid < 128) {
    const float* wrow = p.W[7] + tid * 608;
    float a = 0.f;
    for (int c = 0; c < 512; c += 4) {
      float4 ww = *(const float4*)&wrow[c];
      float4 gg = lds4(F_GV + c);
      a += ww.x * gg.x + ww.y * gg.y + ww.z * gg.z + ww.w * gg.w;
    }
    stsf(F_GO + tid, a);
  }
  __syncthreads();

  // ---- W7: h1[n][132] = lrelu(s7*(W7[:,512:]@[x1;x2;x3] + go) + t7) ------
  {
    const float* W7 = p.W[7];
    const float* s7 = p.s[7];
    const float* t7 = p.t[7];
    int tm = 0, tn = wave;
    while (tn >= 7) { tn -= 7; ++tm; }
    for (int t = wave; t < 56; t += NWAVES) {
      const int mo = tm << 4, no = tn << 4;
      const float* arow = W7 + (mo + lo) * 608 + 512 + khi;
      const int bb = F_XST + (no + lo) * 100 + khi;
      v8f acc = gemm_tile<24>([&](int k) { return *(const v2f*)&arow[k]; },
                              [&](int k) { return lds2(bb + k); });
      const int ob = mo + (hi << 3);
      float4 ga = lds4(F_GO + ob), gb = lds4(F_GO + ob + 4);
      v8f ac2 = {acc[0] + ga.x, acc[1] + ga.y, acc[2] + ga.z, acc[3] + ga.w,
                 acc[4] + gb.x, acc[5] + gb.y, acc[6] + gb.z, acc[7] + gb.w};
      st_tile_bn(F_SCR + (no + lo) * 132 + ob, ob, s7, t7, ac2);
      tn += NWAVES;
      while (tn >= 7) { tn -= 7; ++tm; }
    }
  }
  __syncthreads();

  // ---- W8: h2[n][36] = lrelu(s8*(W8 @ h1)+t8)  (reuse u region) ----------
  {
    const float* W8 = p.W[8];
    const float* s8 = p.s[8];
    const float* t8 = p.t[8];
    int tm = 0, tn = wave;
    while (tn >= 7) { tn -= 7; ++tm; }
    for (int t = wave; t < 14; t += NWAVES) {
      const int mo = tm << 4, no = tn << 4;
      const float* arow = W8 + (mo + lo) * 128 + khi;
      const int bb = F_SCR + (no + lo) * 132 + khi;
      v8f acc = gemm_tile<32>([&](int k) { return *(const v2f*)&arow[k]; },
                              [&](int k) { return lds2(bb + k); });
      const int ob = mo + (hi << 3);
      st_tile_bn(F_U + (no + lo) * 36 + ob, ob, s8, t8, acc);
      tn += NWAVES;
      while (tn >= 7) { tn -= 7; ++tm; }
    }
  }
  __syncthreads();

  // ---- W9: out[n] = W9 @ h2[:, n] ----------------------------------------
  if (tid < NPTS) {
    const float* W9 = p.W[9];
    float a = 0.f;
#pragma unroll
    for (int c = 0; c < 32; c += 4) {
      float4 hh = lds4(F_U + tid * 36 + c);
      float4 ww = *(const float4*)&W9[c];
      a += hh.x * ww.x + hh.y * ww.y + hh.z * ww.z + hh.w * ww.w;
    }
    p.out[(size_t)b * NPTS + tid] = a;
  }
}

extern "C" void kernel_launch(void* const* d_in, const int* in_sizes, int n_in,
                              void* d_out, int out_size, void* d_ws,
                              size_t ws_size, hipStream_t stream) {
  Params p;
  p.obs = (const float*)d_in[0];
  for (int i = 0; i < 10; ++i) p.W[i] = (const float*)d_in[1 + i];
  // setup_inputs dict order interleaves: s0,t0,s1,t1,...
  for (int i = 0; i < 9; ++i) {
    p.s[i] = (const float*)d_in[11 + 2 * i];
    p.t[i] = (const float*)d_in[12 + 2 * i];
  }
  p.out = (float*)d_out;
  (void)hipFuncSetAttribute((const void*)dgcnn_kernel,
                            hipFuncAttributeMaxDynamicSharedMemorySize,
                            SMEM_BYTES);
  dgcnn_kernel<<<1024, NTHREADS, SMEM_BYTES, stream>>>(p);
  (void)in_sizes; (void)n_in; (void)d_ws; (void)ws_size; (void)out_size;
}